// MultichannelMultiheadAttention_87136296501752
// MI455X (gfx1250) — compile-verified
//
#include <hip/hip_runtime.h>
#include <hip/hip_bf16.h>

typedef __attribute__((ext_vector_type(2))) float fx2;
typedef __attribute__((ext_vector_type(4))) float fx4;
typedef __attribute__((ext_vector_type(8))) float fx8;

#define B_ 2
#define C_ 8
#define F_ 256
#define W_ 1024
#define H_ 8
#define HD_ 32

__device__ __forceinline__ fx8 wmma_f32(fx2 a, fx2 b, fx8 c) {
  // V_WMMA_F32_16X16X4_F32 : D(16x16,f32) = A(16x4,f32) x B(4x16,f32) + C
  return __builtin_amdgcn_wmma_f32_16x16x4_f32(false, a, false, b, (short)0, c,
                                               false, false);
}

// -------------------------------------------------------------------------
// Per-channel pointwise projection:  Y[bc] (256x1024) = Wt[c] (256x256) * X[bc]
// Block tile 64(f) x 128(w); 8 waves as 2x4, each wave owns 32x32 (4 WMMA Ds).
// A kept row-major [f][k]; B stored TRANSPOSED [w][k] so every fragment is a
// contiguous 8B-aligned pair -> ds_load_b64, no repacking movs.
// -------------------------------------------------------------------------
__global__ __launch_bounds__(256) void gemm_pc_kernel(
    const float* __restrict__ pw,   // [C,F,F]
    const float* __restrict__ X,    // [B*C,F,W]
    float* __restrict__ Y)          // [B*C,F,W]
{
  const int bc = blockIdx.z;          // 0..15
  const int c  = bc & (C_ - 1);
  const int f0 = blockIdx.y * 64;
  const int w0 = blockIdx.x * 128;

  const float* A  = pw + (size_t)c * F_ * F_;
  const float* Bm = X + (size_t)bc * F_ * W_;
  float*       D  = Y + (size_t)bc * F_ * W_;

  __shared__ float As[64][68];     // [f][k]
  __shared__ float BsT[128][68];   // [w][k]  (transposed)

  const int tid  = threadIdx.x;
  const int lane = tid & 31;
  const int wave = tid >> 5;            // 0..7
  const int wf   = (wave >> 2) * 32;    // f sub-tile base (0/32)
  const int ww   = (wave & 3) * 32;     // w sub-tile base (0/32/64/96)
  const int l16  = lane & 15;
  const int half = lane >> 4;

  fx8 acc00 = {}, acc01 = {}, acc10 = {}, acc11 = {};

  for (int kc = 0; kc < F_; kc += 64) {
    // --- A chunk: 64 rows(f) x 64 cols(k) = 1024 fx4, 4 per thread
#pragma unroll
    for (int r = 0; r < 4; ++r) {
      int idx  = tid + r * 256;         // 0..1023
      int row  = idx >> 4;              // 0..63
      int col4 = (idx & 15) * 4;        // 0..60
      fx4 av = *(const fx4*)(A + (size_t)(f0 + row) * F_ + kc + col4);
      *(fx4*)&As[row][col4] = av;
    }
    // --- B chunk: 64 rows(k) x 128 cols(w) = 2048 fx4, 8 per thread,
    //     scattered transposed into BsT[w][k]
#pragma unroll
    for (int r = 0; r < 8; ++r) {
      int idx   = tid + r * 256;        // 0..2047
      int krow  = idx >> 5;             // 0..63
      int wcol4 = (idx & 31) * 4;       // 0..124
      fx4 bv = *(const fx4*)(Bm + (size_t)(kc + krow) * W_ + w0 + wcol4);
      BsT[wcol4 + 0][krow] = bv.x;
      BsT[wcol4 + 1][krow] = bv.y;
      BsT[wcol4 + 2][krow] = bv.z;
      BsT[wcol4 + 3][krow] = bv.w;
    }
    __syncthreads();

#pragma unroll
    for (int kk = 0; kk < 64; kk += 4) {
      const int ka = kk + half * 2;
      fx2 a0 = { As[wf + l16][ka],       As[wf + l16][ka + 1] };
      fx2 a1 = { As[wf + 16 + l16][ka],  As[wf + 16 + l16][ka + 1] };
      fx2 b0 = { BsT[ww + l16][ka],      BsT[ww + l16][ka + 1] };
      fx2 b1 = { BsT[ww + 16 + l16][ka], BsT[ww + 16 + l16][ka + 1] };
      acc00 = wmma_f32(a0, b0, acc00);
      acc01 = wmma_f32(a0, b1, acc01);
      acc10 = wmma_f32(a1, b0, acc10);
      acc11 = wmma_f32(a1, b1, acc11);
    }
    __syncthreads();
  }

#pragma unroll
  for (int j = 0; j < 8; ++j) {
    const int fA = f0 + wf + j + half * 8;
    const int fB = fA + 16;
    const int wA = w0 + ww + l16;
    D[(size_t)fA * W_ + wA]      = acc00[j];
    D[(size_t)fA * W_ + wA + 16] = acc01[j];
    D[(size_t)fB * W_ + wA]      = acc10[j];
    D[(size_t)fB * W_ + wA + 16] = acc11[j];
  }
}

// -------------------------------------------------------------------------
// Cross-channel 1x9 conv (pad 4) + head split + frame-norm + rotary.
// One thread owns (b, o, head, w): computes all HD=32 values in registers.
// Output layout: [ (b*C+o)*H + head ][ w ][ d ]
// -------------------------------------------------------------------------
__global__ __launch_bounds__(256) void conv_norm_rope_kernel(
    const float* __restrict__ P,    // [B*C,F,W] projection output
    const float* __restrict__ cw,   // [C,C,1,9]
    const float* __restrict__ gam,  // [HD]
    const float* __restrict__ bet,  // [HD]
    float* __restrict__ O,          // [B*C*H, W, HD]
    int do_norm_rope)
{
  const int gid = blockIdx.x * blockDim.x + threadIdx.x;  // B*C*H*W = 131072
  const int w   = gid & (W_ - 1);
  const int t1  = gid >> 10;            // 0..127  (= bco*H + h)
  const int h   = t1 & (H_ - 1);
  const int bco = t1 >> 3;              // 0..15
  const int b   = bco >> 3;
  const int o   = bco & (C_ - 1);

  float acc[HD_];
#pragma unroll
  for (int d = 0; d < HD_; ++d) acc[d] = 0.0f;

  const int t_lo = (w < 4) ? (4 - w) : 0;
  const int t_hi = (w > W_ - 5) ? (W_ + 4 - w) : 9;

  for (int i = 0; i < C_; ++i) {
    const float* Pi = P + ((size_t)(b * C_ + i) * F_ + h * HD_) * W_;
    const float* wv = cw + (size_t)(o * C_ + i) * 9;
    float wreg[9];
#pragma unroll
    for (int t = 0; t < 9; ++t) wreg[t] = wv[t];
    for (int t = t_lo; t < t_hi; ++t) {
      const float cv = wreg[t];
      const int ws = w + t - 4;
#pragma unroll
      for (int d = 0; d < HD_; ++d)
        acc[d] = fmaf(Pi[(size_t)d * W_ + ws], cv, acc[d]);
    }
  }

  if (do_norm_rope) {
    float m = 0.0f;
#pragma unroll
    for (int d = 0; d < HD_; ++d) m += acc[d];
    m *= (1.0f / HD_);
    float var = 0.0f;
#pragma unroll
    for (int d = 0; d < HD_; ++d) { float t = acc[d] - m; var = fmaf(t, t, var); }
    var *= (1.0f / HD_);
    const float r = rsqrtf(var + 1e-5f);
#pragma unroll
    for (int d = 0; d < HD_; ++d)
      acc[d] = (acc[d] - m) * r * gam[d] + bet[d];
    // rotary on first RD=16 dims (8 interleaved pairs)
#pragma unroll
    for (int i = 0; i < 8; ++i) {
      // inv = 10000^(-i/8) = exp(-i * ln(10000)/8)
      const float inv = expf(-(float)i * 1.1512925464970229f);
      const float ang = (float)w * inv;
      const float cs = cosf(ang);
      const float sn = sinf(ang);
      const float x1 = acc[2 * i];
      const float x2 = acc[2 * i + 1];
      acc[2 * i]     = x1 * cs - x2 * sn;
      acc[2 * i + 1] = x2 * cs + x1 * sn;
    }
  }

  float* op = O + ((size_t)t1 * W_ + w) * HD_;
#pragma unroll
  for (int d4 = 0; d4 < HD_; d4 += 4) {
    fx4 v = { acc[d4], acc[d4 + 1], acc[d4 + 2], acc[d4 + 3] };
    *(fx4*)(op + d4) = v;
  }
}

// -------------------------------------------------------------------------
// Flash-style attention per (b,c,head).  One wave = one 16-row q tile,
// 8 waves/block share K/V tiles via LDS.  QK^T and PV both use fp32 WMMA.
// V stored transposed in LDS so PV B-fragments are contiguous pairs.
// Output written in bcfw layout for the output projection GEMM.
// -------------------------------------------------------------------------
__global__ __launch_bounds__(256) void attn_kernel(
    const float* __restrict__ Q,    // [BCH, W, HD]
    const float* __restrict__ K,
    const float* __restrict__ V,
    float* __restrict__ Aout)       // [B*C, F, W]
{
  const int head = blockIdx.y;            // 0..127 = bc*H + h
  const int qblk = blockIdx.x;            // 0..7
  const int bc = head >> 3;
  const int h  = head & (H_ - 1);

  const int tid  = threadIdx.x;
  const int lane = tid & 31;
  const int wave = tid >> 5;
  const int l16  = lane & 15;
  const int half = lane >> 4;

  const float* Qh = Q + (size_t)head * W_ * HD_;
  const float* Kh = K + (size_t)head * W_ * HD_;
  const float* Vh = V + (size_t)head * W_ * HD_;

  __shared__ float Ks[16][34];   // [kcol][d]  (QK^T B-frag pairs contiguous)
  __shared__ float VsT[32][18];  // [d][krow]  (PV   B-frag pairs contiguous)
  __shared__ float Ps[8][16][18];

  // Q tile for this wave (rows q0..q0+15), pre-scaled by 1/sqrt(F)=1/16,
  // held as the 8 A-fragments for the K-dim sweep (d = 0..31 step 4).
  const int q0 = qblk * 128 + wave * 16;
  fx2 qf[8];
#pragma unroll
  for (int s = 0; s < 8; ++s) {
    const int d = 4 * s + half * 2;
    const float* qp = Qh + (size_t)(q0 + l16) * HD_ + d;
    qf[s].x = qp[0] * 0.0625f;
    qf[s].y = qp[1] * 0.0625f;
  }

  fx8 o0 = {};
  fx8 o1 = {};
  float mrow[8], lrow[8];
#pragma unroll
  for (int j = 0; j < 8; ++j) { mrow[j] = -3.0e38f; lrow[j] = 0.0f; }

  for (int kt = 0; kt < W_; kt += 16) {
    __syncthreads();
    // cooperative load of K/V tiles: 16 rows x 32 dims each (contiguous 512f)
    for (int idx = tid; idx < 512; idx += 256) {
      const int r = idx >> 5, cc = idx & 31;
      Ks[r][cc]  = Kh[(size_t)kt * HD_ + idx];
      VsT[cc][r] = Vh[(size_t)kt * HD_ + idx];   // transposed scatter
    }
    __syncthreads();

    // S = Q * K^T  (16x16), K-dim = HD in steps of 4
    fx8 s = {};
#pragma unroll
    for (int st = 0; st < 8; ++st) {
      const int d = 4 * st + half * 2;
      fx2 bfrag = { Ks[l16][d], Ks[l16][d + 1] };   // B[d][n]=K[n][d]
      s = wmma_f32(qf[st], bfrag, s);
    }

    // online softmax; row M = j + 8*half lives across 16 lanes of this half
#pragma unroll
    for (int j = 0; j < 8; ++j) {
      float v = s[j];
      float rm = v;
#pragma unroll
      for (int msk = 1; msk < 16; msk <<= 1)
        rm = fmaxf(rm, __shfl_xor(rm, msk, 32));
      const float mnew = fmaxf(mrow[j], rm);
      const float corr = expf(mrow[j] - mnew);
      const float p = expf(v - mnew);
      float rs = p;
#pragma unroll
      for (int msk = 1; msk < 16; msk <<= 1)
        rs += __shfl_xor(rs, msk, 32);
      lrow[j] = lrow[j] * corr + rs;
      mrow[j] = mnew;
      o0[j] *= corr;
      o1[j] *= corr;
      Ps[wave][j + 8 * half][l16] = p;   // wave-private, LDS ops in-order
    }

    // O += P (16x16) * V (16x32)
#pragma unroll
    for (int st = 0; st < 4; ++st) {
      const int kk = 4 * st + half * 2;
      fx2 pa  = { Ps[wave][l16][kk],  Ps[wave][l16][kk + 1] };
      fx2 vb0 = { VsT[l16][kk],       VsT[l16][kk + 1] };
      fx2 vb1 = { VsT[16 + l16][kk],  VsT[16 + l16][kk + 1] };
      o0 = wmma_f32(pa, vb0, o0);
      o1 = wmma_f32(pa, vb1, o1);
    }
  }

  // epilogue: normalize rows, scatter to bcfw layout
#pragma unroll
  for (int j = 0; j < 8; ++j) {
    const float inv = 1.0f / lrow[j];
    const int wq = q0 + j + 8 * half;            // q row -> w index
    const int fq = h * HD_ + l16;                // v-dim -> f index
    Aout[((size_t)bc * F_ + fq) * W_ + wq]      = o0[j] * inv;
    Aout[((size_t)bc * F_ + fq + 16) * W_ + wq] = o1[j] * inv;
  }
}

// -------------------------------------------------------------------------
// Final 8x8 channel mix: out[b,d,f,w] = sum_c y[b,c,f,w] * dw[d,c]
// -------------------------------------------------------------------------
__global__ __launch_bounds__(256) void mix_kernel(
    const float* __restrict__ Yin, const float* __restrict__ dw,
    float* __restrict__ out)
{
  const int idx = blockIdx.x * 256 + threadIdx.x;   // B*F*W = 524288
  const int w = idx & (W_ - 1);
  const int f = (idx >> 10) & (F_ - 1);
  const int b = idx >> 18;

  float xin[C_];
#pragma unroll
  for (int c = 0; c < C_; ++c)
    xin[c] = Yin[((size_t)(b * C_ + c) * F_ + f) * W_ + w];
#pragma unroll
  for (int d = 0; d < C_; ++d) {
    float a = 0.0f;
#pragma unroll
    for (int c = 0; c < C_; ++c) a = fmaf(xin[c], dw[d * C_ + c], a);
    out[((size_t)(b * C_ + d) * F_ + f) * W_ + w] = a;
  }
}

// -------------------------------------------------------------------------
extern "C" void kernel_launch(void* const* d_in, const int* in_sizes, int n_in,
                              void* d_out, int out_size, void* d_ws,
                              size_t ws_size, hipStream_t stream) {
  (void)in_sizes; (void)n_in; (void)out_size; (void)ws_size;
  const float* x      = (const float*)d_in[0];
  const float* q_pw   = (const float*)d_in[1];
  const float* q_cw   = (const float*)d_in[2];
  const float* qn_g   = (const float*)d_in[3];
  const float* qn_b   = (const float*)d_in[4];
  const float* k_pw   = (const float*)d_in[5];
  const float* k_cw   = (const float*)d_in[6];
  const float* kn_g   = (const float*)d_in[7];
  const float* kn_b   = (const float*)d_in[8];
  const float* v_pw   = (const float*)d_in[9];
  const float* v_cw   = (const float*)d_in[10];
  const float* out_pw = (const float*)d_in[11];
  const float* out_dw = (const float*)d_in[12];
  float* out = (float*)d_out;

  const size_t N = (size_t)B_ * C_ * F_ * W_;   // 4M floats per buffer
  float* buf = (float*)d_ws;                    // needs 6*N*4 = 96 MB
  float* qp = buf + 0 * N;
  float* kp = buf + 1 * N;
  float* vp = buf + 2 * N;
  float* qh = buf + 3 * N;
  float* kh = buf + 4 * N;
  float* vh = buf + 5 * N;
  float* attn = qp;   // qp dead after conv stage
  float* y    = kp;   // kp dead after conv stage

  dim3 gg(W_ / 128, F_ / 64, B_ * C_);          // (8,4,16)
  gemm_pc_kernel<<<gg, 256, 0, stream>>>(q_pw, x, qp);
  gemm_pc_kernel<<<gg, 256, 0, stream>>>(k_pw, x, kp);
  gemm_pc_kernel<<<gg, 256, 0, stream>>>(v_pw, x, vp);

  const int convBlocks = (B_ * C_ * H_ * W_) / 256;   // 512
  conv_norm_rope_kernel<<<convBlocks, 256, 0, stream>>>(qp, q_cw, qn_g, qn_b, qh, 1);
  conv_norm_rope_kernel<<<convBlocks, 256, 0, stream>>>(kp, k_cw, kn_g, kn_b, kh, 1);
  conv_norm_rope_kernel<<<convBlocks, 256, 0, stream>>>(vp, v_cw, qn_g, qn_b, vh, 0);

  attn_kernel<<<dim3(8, B_ * C_ * H_), 256, 0, stream>>>(qh, kh, vh, attn);

  gemm_pc_kernel<<<gg, 256, 0, stream>>>(out_pw, attn, y);

  mix_kernel<<<(B_ * F_ * W_) / 256, 256, 0, stream>>>(y, out_dw, out);
}